// HoltsWintersNoTrend_12575664243192
// MI455X (gfx1250) — compile-verified
//
#include <hip/hip_runtime.h>
#include <stdint.h>

// ---------------------------------------------------------------------------
// Holt-Winters (no trend). B=4096 rows, T=4096 steps, output = last 14
// smoothed values per row (row-major [4096,14] f32).
//
// smooth_t = a*(x_t/s_t) + r*smooth_{t-1},  r = 1-a
//          = r^{t+1} smooth_0 + sum_{j<=t} r^{t-j} c_j,  c_j = a*x_j/s_j
//
// HBM-bandwidth-bound scan (~2 VALU ops/element):
//  * one wave per row; stream the row tail through an 8-deep LDS ring with
//    global_load_async_to_lds_b128 (ASYNCcnt) -> deep prefetch, zero VGPR cost
//  * steady state: 2x ds_load_b128 + 4-fma chain per 128 floats, per-lane
//    weighted accumulation (no cross-lane ops until the final reduce)
//  * exact 32-step cross-lane prefix scan for the last chunk (the outputs)
//  * runtime truncation: a=0.1 => r^k == 0 in fp32 for k >~ 1000; read only
//    the last 2048 steps (32MB @ 23.3TB/s ~ 1.4us); full T if r is large
// ---------------------------------------------------------------------------

#define B_ROWS 4096
#define T_LEN  4096
#define SLEN   7
#define NPREDS 14

#define WAVES_PER_BLOCK 8
#define RING   8      // outer chunks in flight per wave
#define GF     128    // floats per outer chunk = 512B = one b128 async / wave

// ---- gfx1250 async-to-LDS helpers (inline asm; portable across toolchains)
__device__ __forceinline__ void async_copy_b128(uint32_t lds_byte_addr,
                                                const float* gsrc) {
#if defined(__HIP_DEVICE_COMPILE__)
  asm volatile("global_load_async_to_lds_b128 %0, %1, off"
               :: "v"(lds_byte_addr), "v"(gsrc)
               : "memory");
#endif
}

#if defined(__HIP_DEVICE_COMPILE__) && __has_builtin(__builtin_amdgcn_s_wait_asynccnt)
#define WAIT_ASYNC(n) do { __builtin_amdgcn_s_wait_asynccnt(n); \
                           asm volatile("" ::: "memory"); } while (0)
#else
#define WAIT_ASYNC(n) asm volatile("s_wait_asynccnt %0" :: "i"(n) : "memory")
#endif

#if defined(__HIP_DEVICE_COMPILE__) && __has_builtin(__builtin_amdgcn_s_wait_dscnt)
#define WAIT_DS0() do { __builtin_amdgcn_s_wait_dscnt(0); \
                        asm volatile("" ::: "memory"); } while (0)
#else
#define WAIT_DS0() asm volatile("s_wait_dscnt 0x0" ::: "memory")
#endif

__global__ __launch_bounds__(WAVES_PER_BLOCK * 32)
void hw_scan_kernel(const float* __restrict__ series,
                    const int*   __restrict__ shifts,
                    const float* __restrict__ alpha_p,
                    const float* __restrict__ init_season,
                    float*       __restrict__ out) {
  __shared__ __align__(16) float ring[WAVES_PER_BLOCK][RING][GF];    // 32 KB
  __shared__ __align__(16) float wtab[WAVES_PER_BLOCK][SLEN][GF];    // 28 KB
  __shared__ float wrs_tab[WAVES_PER_BLOCK][SLEN][32];               //  7 KB
  __shared__ float aS_tab[WAVES_PER_BLOCK][8];   // alpha/S(phase), per wave

  const int lane = threadIdx.x & 31;          // wave32
  const int wv   = threadIdx.x >> 5;
  const int row  = blockIdx.x * WAVES_PER_BLOCK + wv;

  const float alpha = alpha_p[0];
  const float r     = 1.0f - alpha;
  const int   shift = shifts[row];
  const float* rowp = series + (size_t)row * T_LEN;

  // Runtime truncation: r <= 0.958 => r^2016 < 1e-37 (invisible in fp32).
  const int nOuter = (r <= 0.958f) ? (T_LEN / GF / 2) : (T_LEN / GF);
  const int t0     = T_LEN - nOuter * GF;

  // --- powers of r: wave-uniform squarings + branchless per-lane selects ---
  const float r2 = r * r, r4 = r2 * r2, r8 = r4 * r4, r16 = r8 * r8;
  const float r32 = r16 * r16, r64 = r32 * r32, r128 = r64 * r64;
  const float rpw[7] = {r, r2, r4, r8, r16, r32, r64};
  auto powr = [&](int e) {           // r^e, 0 <= e <= 127, branchless
    float p = 1.0f;
#pragma unroll
    for (int i = 0; i < 7; ++i) p *= ((e >> i) & 1) ? rpw[i] : 1.0f;
    return p;
  };
  const float ir = 1.0f / r;
  const float i2 = ir * ir, i4 = i2 * i2, i8 = i4 * i4, i16 = i8 * i8;
  const float ipw[5] = {ir, i2, i4, i8, i16};

  const float w    = powr(31 - lane);          // r^(31-lane)
  const float rpow = powr(lane);               // r^lane
  float rinv = 1.0f;                           // r^-lane
#pragma unroll
  for (int i = 0; i < 5; ++i) rinv *= ((lane >> i) & 1) ? ipw[i] : 1.0f;

  // --- per-phase alpha/S(p): S(p) = init_season[(p - shift) mod 7] ---
  if (lane < SLEN) {
    int si = lane - shift; if (si < 0) si += SLEN;
    aS_tab[wv][lane] = alpha / init_season[si];
  }
  asm volatile("" ::: "memory");   // same-wave LDS write -> read ordering

  // --- per-lane season weight tables (phase kept incrementally, no mods) ---
  // fast path (128/chunk): element m=4*lane+j of chunk k, t = t0+128k+m,
  //   (128k)%7 == (2*(k%7))%7  ->  wtab[k%7][m] = r^(127-m) * a / S(...)
  {
    float rp = powr(127 - 4 * lane);           // r^(127-m) for j=0
    int phj = (t0 + 4 * lane) % SLEN;          // phase at cyc=0, j=0
#pragma unroll
    for (int j = 0; j < 4; ++j) {
      const int m = 4 * lane + j;
      int ph = phj;
#pragma unroll 1
      for (int cyc = 0; cyc < SLEN; ++cyc) {
        wtab[wv][cyc][m] = rp * aS_tab[wv][ph];
        ph += 2; if (ph >= SLEN) ph -= SLEN;
      }
      rp *= ir;                                // next j: r^(127-m-1)
      phj += 1; if (phj >= SLEN) phj -= SLEN;
    }
  }
  // scalar path (32/sub): sub s, t = t0+32s+lane, (32s)%7 == (4*(s%7))%7
  {
    int ph = (t0 + lane) % SLEN;
#pragma unroll 1
    for (int cyc = 0; cyc < SLEN; ++cyc) {
      wrs_tab[wv][cyc][lane] = w * aS_tab[wv][ph];
      ph += 4; if (ph >= SLEN) ph -= SLEN;
    }
  }

  // --- prologue: fill the async ring ---
  for (int k = 0; k < RING; ++k) {
    uint32_t ldst = (uint32_t)(uintptr_t)&ring[wv][k & (RING - 1)][4 * lane];
    async_copy_b128(ldst, rowp + t0 + k * GF + 4 * lane);
  }

  // seed r^{t0}*smooth_0 (non-zero only when processing the full series)
  float acc = (t0 == 0 && lane == 0) ? rowp[0] : 0.0f;
  int cycK = 0;  // k % 7

  // invariant after processing values up to time tau (any granularity):
  //   sum_over_lanes(acc) == sum_{t<=tau} r^{tau-t} c_t
  auto consume_fast = [&](int k) {
    const float* buf = &ring[wv][k & (RING - 1)][0];
    float4 x  = *(const float4*)(buf + 4 * lane);
    float4 wt = *(const float4*)(&wtab[wv][cycK][4 * lane]);
    float t = x.x * wt.x;
    t = fmaf(x.y, wt.y, t);
    t = fmaf(x.z, wt.z, t);
    t = fmaf(x.w, wt.w, t);
    acc = fmaf(acc, r128, t);
    cycK += 1; if (cycK == SLEN) cycK = 0;
  };

  auto consume_last = [&](int k) {   // final chunk: 3 scalar subs + prefix
    const float* buf = &ring[wv][k & (RING - 1)][0];
    int cyc = (4 * k) % SLEN;        // sub index s = 4k
#pragma unroll
    for (int j = 0; j < 3; ++j) {
      float x   = buf[j * 32 + lane];
      float wrs = wrs_tab[wv][cyc][lane];
      acc = fmaf(acc, r32, x * wrs);
      cyc += 4; if (cyc >= SLEN) cyc -= SLEN;
    }
    // smooth_prev at t = T-33: reduce per-lane accumulators
    float sp = acc;
    for (int d = 16; d >= 1; d >>= 1) sp += __shfl_xor(sp, d, 32);
    // exact last 32 steps: t = (T-32) + lane
    float x  = buf[3 * 32 + lane];
    int   ph = ((T_LEN - 32) + lane) % SLEN;
    float aS = aS_tab[wv][ph];                 // alpha / s_t
    float c  = x * aS;
    // inclusive cross-lane prefix of u = c * r^-lane
    float P = c * rinv;
    for (int d = 1; d < 32; d <<= 1) {
      float o = __shfl_up(P, (unsigned)d, 32);
      if (lane >= d) P += o;
    }
    // smooth_t = r^{lane+1}*smooth_prev + r^lane * P ; out = smooth_t * s_t
    float sm  = fmaf(sp, rpow * r, P * rpow);
    float res = sm * (alpha / aS);             // s_t = alpha / (alpha/s_t)
    if (lane >= 32 - NPREDS)
      out[(size_t)row * NPREDS + (lane - (32 - NPREDS))] = res;
  };

  // --- steady state: wait oldest, consume, refill same slot ---
  for (int k = 0; k < nOuter - RING; ++k) {
    WAIT_ASYNC(RING - 1);
    consume_fast(k);
    WAIT_DS0();  // ring reads retired before async rewrite of this slot
    uint32_t ldst = (uint32_t)(uintptr_t)&ring[wv][k & (RING - 1)][4 * lane];
    async_copy_b128(ldst, rowp + t0 + (k + RING) * GF + 4 * lane);
  }

  // --- epilogue: drain the ring with constant wait immediates ---
  {
    const int k0 = nOuter - RING;
#define EPI(J) do { WAIT_ASYNC(RING - 1 - (J)); consume_fast(k0 + (J)); } while (0)
    EPI(0); EPI(1); EPI(2); EPI(3); EPI(4); EPI(5); EPI(6);
#undef EPI
    WAIT_ASYNC(0);
    consume_last(k0 + 7);            // k0+7 == nOuter-1: the last chunk
  }
}

extern "C" void kernel_launch(void* const* d_in, const int* in_sizes, int n_in,
                              void* d_out, int out_size, void* d_ws,
                              size_t ws_size, hipStream_t stream) {
  (void)in_sizes; (void)n_in; (void)out_size; (void)d_ws; (void)ws_size;
  const float* series      = (const float*)d_in[0];
  const int*   shifts      = (const int*)d_in[1];
  const float* alpha       = (const float*)d_in[2];
  // d_in[3] = gamma (unused by the reference), d_in[5] = n_preds (==14)
  const float* init_season = (const float*)d_in[4];
  float* out = (float*)d_out;

  dim3 grid(B_ROWS / WAVES_PER_BLOCK);
  dim3 block(WAVES_PER_BLOCK * 32);
  hw_scan_kernel<<<grid, block, 0, stream>>>(series, shifts, alpha,
                                             init_season, out);
}